// AttentionBlock_13846974562895
// MI455X (gfx1250) — compile-verified
//
#include <hip/hip_runtime.h>
#include <math.h>

// ---------------------------------------------------------------------------
// AttentionBlock for MI455X (gfx1250, wave32, WMMA).
// FP32 end-to-end using V_WMMA_F32_16X16X4_F32 (exact fp32 matrix math).
// - LDS double-buffered (1 barrier/chunk)
// - k-pair-major float2 B layout: B fragment = one contiguous ds_load_b64
// - A tiles staged with CDNA5 async DMA: global_load_async_to_lds_b128
//   (ASYNCcnt-tracked, no VGPR round trip), synced via s_wait_asynccnt.
// Workspace layout (floats): q[8M] | k[8M] | v[8M] | scores[33.5M]  (~224 MB)
// ---------------------------------------------------------------------------

typedef __attribute__((ext_vector_type(2))) float v2f;
typedef __attribute__((ext_vector_type(4))) float f4;
typedef __attribute__((ext_vector_type(8))) float v8f;

#define B_ 8
#define T_ 2048
#define C_ 512
#define K_ 512
#define OUTW_ (C_ + K_)   // 1024

#define BNP_ 72           // padded row length (in float2) of B tiles

__device__ __forceinline__ v8f wmma_f32(v2f a, v2f b, v8f c) {
  // D = A(16x4) x B(4x16) + C(16x16), all fp32
  return __builtin_amdgcn_wmma_f32_16x16x4_f32(
      /*neg_a=*/false, a, /*neg_b=*/false, b,
      /*c_mod=*/(short)0, c, /*reuse_a=*/false, /*reuse_b=*/false);
}

// Async DMA: copy 16 bytes global -> LDS for this lane (ASYNCcnt-tracked).
__device__ __forceinline__ void async_copy16(void* lds_dst, const float* gsrc) {
  const unsigned lds_off = (unsigned)(uintptr_t)lds_dst;   // LDS aperture: addr[31:0]
  const unsigned long long ga = (unsigned long long)(uintptr_t)gsrc;
  asm volatile("global_load_async_to_lds_b128 %0, %1, off"
               :
               : "v"(lds_off), "v"(ga)
               : "memory");
}

__device__ __forceinline__ void wait_async0() {
  asm volatile("s_wait_asynccnt 0x0" ::: "memory");
}

// One BK=16 chunk of wave-level MMA: wave computes two 16x16 tiles.
// As: [64][20] floats (row-major, padded).  Bs: [8][BNP_] float2,
// Bs[kp][n] = {B[2kp][n], B[2kp+1][n]}  -> fragment is one b64 load.
__device__ __forceinline__ void mma_step(const float (&As)[64][20],
                                         const v2f (&Bs)[8][BNP_],
                                         int wm, int wn, int laneM, int hi,
                                         v8f& acc0, v8f& acc1) {
  const float* ar = &As[wm * 16 + laneM][0];
#pragma unroll
  for (int kk = 0; kk < 16; kk += 4) {
    const int kb = kk + 2 * hi;   // lanes 0-15: K=kk,kk+1 ; lanes 16-31: K=kk+2,kk+3
    v2f a;
    a.x = ar[kb];
    a.y = ar[kb + 1];
    const v2f b0 = Bs[kb >> 1][wn * 16 + laneM];
    const v2f b1 = Bs[kb >> 1][wn * 16 + 16 + laneM];
    acc0 = wmma_f32(a, b0, acc0);
    acc1 = wmma_f32(a, b1, acc1);
  }
}

// ---------------------------------------------------------------------------
// Kernel 1: fused QKV projection.  out[m,n] = sum_c x[m,c]*W[n,c] + b[n]
// grid = (M/64, K_/64, 3), block = 256
// ---------------------------------------------------------------------------
__global__ __launch_bounds__(256) void qkv_proj_kernel(
    const float* __restrict__ x,
    const float* __restrict__ Wq, const float* __restrict__ bq,
    const float* __restrict__ Wk, const float* __restrict__ bk,
    const float* __restrict__ Wv, const float* __restrict__ bv,
    float* __restrict__ qkv_ws) {
  __shared__ float As[2][64][20];
  __shared__ v2f Bs[2][8][BNP_];

  const int tid = threadIdx.x;
  const int lane = tid & 31, wave = tid >> 5;
  const int laneM = lane & 15, hi = lane >> 4;
  const int wm = wave & 3, wn = (wave >> 2) << 1;

  const int proj = blockIdx.z;
  const float* W = (proj == 0) ? Wq : (proj == 1) ? Wk : Wv;
  const float* bias = (proj == 0) ? bq : (proj == 1) ? bk : bv;
  float* out = qkv_ws + (size_t)proj * ((size_t)B_ * T_ * K_);

  const int m0 = blockIdx.x * 64;
  const int n0 = blockIdx.y * 64;

  v8f acc0 = {0.f, 0.f, 0.f, 0.f, 0.f, 0.f, 0.f, 0.f};
  v8f acc1 = {0.f, 0.f, 0.f, 0.f, 0.f, 0.f, 0.f, 0.f};

  const int lr = tid >> 2;         // 0..63
  const int lk = (tid & 3) << 2;   // 0,4,8,12
  const float* ap = x + (size_t)(m0 + lr) * C_ + lk;
  const float* bp = W + (size_t)(n0 + lr) * C_ + lk;

  // prologue: stage chunk 0 into buffer 0
  {
    async_copy16(&As[0][lr][lk], ap);             // A: async DMA to LDS
    f4 wv = *(const f4*)(bp);                     // B: transpose via regs
    Bs[0][(lk >> 1) + 0][lr] = (v2f){wv.x, wv.y};
    Bs[0][(lk >> 1) + 1][lr] = (v2f){wv.z, wv.w};
  }
  wait_async0();
  __syncthreads();

  const int nch = C_ / 16;   // 32 chunks
  for (int c = 0; c < nch; ++c) {
    const int cur = c & 1;
    const bool more = (c + 1) < nch;
    f4 nb;
    if (more) {
      async_copy16(&As[cur ^ 1][lr][lk], ap + (c + 1) * 16);  // overlaps MMAs
      nb = *(const f4*)(bp + (c + 1) * 16);
    }
    mma_step(As[cur], Bs[cur], wm, wn, laneM, hi, acc0, acc1);
    if (more) {
      Bs[cur ^ 1][(lk >> 1) + 0][lr] = (v2f){nb.x, nb.y};
      Bs[cur ^ 1][(lk >> 1) + 1][lr] = (v2f){nb.z, nb.w};
    }
    wait_async0();
    __syncthreads();
  }

  const int row = m0 + wm * 16 + hi * 8;
  const int c0 = n0 + wn * 16 + laneM;
  const int c1 = c0 + 16;
  const float b0v = bias[c0], b1v = bias[c1];
#pragma unroll
  for (int r = 0; r < 8; ++r) {
    out[(size_t)(row + r) * K_ + c0] = acc0[r] + b0v;
    out[(size_t)(row + r) * K_ + c1] = acc1[r] + b1v;
  }
}

// ---------------------------------------------------------------------------
// Kernel 2: scores[b,i,j] = (j>i) ? -inf : dot(q[b,i,:], k[b,j,:]) / sqrt(512)
// grid = (T/64, T/64, B), block = 256.  Fully-masked tiles early-out.
// ---------------------------------------------------------------------------
__global__ __launch_bounds__(256) void scores_kernel(
    const float* __restrict__ q, const float* __restrict__ kmat,
    float* __restrict__ scores) {
  const int b = blockIdx.z;
  const int m0 = blockIdx.x * 64;
  const int n0 = blockIdx.y * 64;
  float* S = scores + (size_t)b * T_ * T_;

  if (n0 > m0 + 63) {  // entire tile is strictly above the diagonal
    const f4 minf = {-INFINITY, -INFINITY, -INFINITY, -INFINITY};
    for (int idx = threadIdx.x; idx < 64 * 16; idx += 256) {
      const int r = idx >> 4, c4 = (idx & 15) << 2;
      *(f4*)(S + (size_t)(m0 + r) * T_ + n0 + c4) = minf;
    }
    return;
  }

  __shared__ float As[2][64][20];
  __shared__ v2f Bs[2][8][BNP_];

  const int tid = threadIdx.x;
  const int lane = tid & 31, wave = tid >> 5;
  const int laneM = lane & 15, hi = lane >> 4;
  const int wm = wave & 3, wn = (wave >> 2) << 1;

  v8f acc0 = {0.f, 0.f, 0.f, 0.f, 0.f, 0.f, 0.f, 0.f};
  v8f acc1 = {0.f, 0.f, 0.f, 0.f, 0.f, 0.f, 0.f, 0.f};

  const int lr = tid >> 2;
  const int lk = (tid & 3) << 2;
  const float* ap = q + (size_t)b * T_ * K_ + (size_t)(m0 + lr) * K_ + lk;
  const float* bp = kmat + (size_t)b * T_ * K_ + (size_t)(n0 + lr) * K_ + lk;

  {
    async_copy16(&As[0][lr][lk], ap);
    f4 kv = *(const f4*)(bp);
    Bs[0][(lk >> 1) + 0][lr] = (v2f){kv.x, kv.y};
    Bs[0][(lk >> 1) + 1][lr] = (v2f){kv.z, kv.w};
  }
  wait_async0();
  __syncthreads();

  const int nch = K_ / 16;
  for (int c = 0; c < nch; ++c) {
    const int cur = c & 1;
    const bool more = (c + 1) < nch;
    f4 nb;
    if (more) {
      async_copy16(&As[cur ^ 1][lr][lk], ap + (c + 1) * 16);
      nb = *(const f4*)(bp + (c + 1) * 16);
    }
    mma_step(As[cur], Bs[cur], wm, wn, laneM, hi, acc0, acc1);
    if (more) {
      Bs[cur ^ 1][(lk >> 1) + 0][lr] = (v2f){nb.x, nb.y};
      Bs[cur ^ 1][(lk >> 1) + 1][lr] = (v2f){nb.z, nb.w};
    }
    wait_async0();
    __syncthreads();
  }

  const float sc = 0.04419417382415922f;  // 1/sqrt(512)
  const int row = m0 + wm * 16 + hi * 8;
  const int c0 = n0 + wn * 16 + laneM;
  const int c1 = c0 + 16;
#pragma unroll
  for (int r = 0; r < 8; ++r) {
    const int i = row + r;
    S[(size_t)i * T_ + c0] = (c0 > i) ? -INFINITY : acc0[r] * sc;
    S[(size_t)i * T_ + c1] = (c1 > i) ? -INFINITY : acc1[r] * sc;
  }
}

// ---------------------------------------------------------------------------
// Kernel 3: softmax over axis=1 (QUERY axis): normalize each column j.
// grid = (T/64, B), block = 256: 64 columns/block, 4 thread-segments/column.
// Masked (-inf) entries become exactly 0.
// ---------------------------------------------------------------------------
__global__ __launch_bounds__(256) void softmax_q_kernel(float* __restrict__ scores) {
  const int b = blockIdx.y;
  const int col = threadIdx.x & 63;
  const int seg = threadIdx.x >> 6;  // 0..3
  const int j = blockIdx.x * 64 + col;
  float* S = scores + (size_t)b * T_ * T_;

  // Online max/sum over this segment's rows.
  float mx = -INFINITY, s = 0.f;
  for (int i = seg; i < T_; i += 4) {
    const float xv = S[(size_t)i * T_ + j];
    if (xv == -INFINITY) continue;           // contributes 0
    if (xv > mx) {
      s = s * expf(mx - xv) + 1.f;           // mx may be -inf: expf(-inf)=0
      mx = xv;
    } else {
      s += expf(xv - mx);
    }
  }

  __shared__ float smx[4][64];
  __shared__ float ssm[4][64];
  __shared__ float gmx[64];
  __shared__ float ginv[64];
  smx[seg][col] = mx;
  ssm[seg][col] = s;
  __syncthreads();

  if (seg == 0) {
    float M = smx[0][col];
#pragma unroll
    for (int t = 1; t < 4; ++t) M = fmaxf(M, smx[t][col]);
    float ss = 0.f;
#pragma unroll
    for (int t = 0; t < 4; ++t) {
      const float mi = smx[t][col];
      ss += (mi == -INFINITY) ? 0.f : ssm[t][col] * expf(mi - M);
    }
    gmx[col] = M;
    ginv[col] = 1.0f / ss;
  }
  __syncthreads();

  const float M = gmx[col], inv = ginv[col];
  for (int i = seg; i < T_; i += 4) {
    const size_t off = (size_t)i * T_ + j;
    const float xv = S[off];
    S[off] = (xv == -INFINITY) ? 0.f : expf(xv - M) * inv;
  }
}

// ---------------------------------------------------------------------------
// Kernel 4: attn = probs @ v, written into out[b, t, 512:1024].
// probs[i,s]==0 for s>i  ->  K loop clamped at m0+64.
// grid = (T/64, K_/64, B), block = 256
// ---------------------------------------------------------------------------
__global__ __launch_bounds__(256) void attn_kernel(
    const float* __restrict__ probs, const float* __restrict__ v,
    float* __restrict__ out) {
  __shared__ float As[2][64][20];
  __shared__ v2f Bs[2][8][BNP_];

  const int tid = threadIdx.x;
  const int lane = tid & 31, wave = tid >> 5;
  const int laneM = lane & 15, hi = lane >> 4;
  const int wm = wave & 3, wn = (wave >> 2) << 1;

  const int b = blockIdx.z;
  const int m0 = blockIdx.x * 64;
  const int n0 = blockIdx.y * 64;

  const float* A = probs + (size_t)b * T_ * T_;
  const float* V = v + (size_t)b * T_ * K_;
  float* O = out + (size_t)b * T_ * OUTW_;

  v8f acc0 = {0.f, 0.f, 0.f, 0.f, 0.f, 0.f, 0.f, 0.f};
  v8f acc1 = {0.f, 0.f, 0.f, 0.f, 0.f, 0.f, 0.f, 0.f};

  const int lr = tid >> 2;          // A staging: 0..63
  const int lk = (tid & 3) << 2;
  const int kp = tid >> 5;          // B staging: k-pair 0..7
  const int nn = (tid & 31) << 1;   // 0..62
  const float* ap = A + (size_t)(m0 + lr) * T_ + lk;
  const float* bp = V + (size_t)(2 * kp) * K_ + n0 + nn;

  // prologue: stage chunk 0
  {
    async_copy16(&As[0][lr][lk], ap);
    v2f r0 = *(const v2f*)(bp);
    v2f r1 = *(const v2f*)(bp + K_);
    Bs[0][kp][nn + 0] = (v2f){r0.x, r1.x};
    Bs[0][kp][nn + 1] = (v2f){r0.y, r1.y};
  }
  wait_async0();
  __syncthreads();

  const int nch = (m0 + 64) / 16;   // causal zero structure of probs
  for (int c = 0; c < nch; ++c) {
    const int cur = c & 1;
    const bool more = (c + 1) < nch;
    v2f n0r, n1r;
    if (more) {
      async_copy16(&As[cur ^ 1][lr][lk], ap + (c + 1) * 16);
      n0r = *(const v2f*)(bp + (size_t)(c + 1) * 16 * K_);
      n1r = *(const v2f*)(bp + (size_t)(c + 1) * 16 * K_ + K_);
    }
    mma_step(As[cur], Bs[cur], wm, wn, laneM, hi, acc0, acc1);
    if (more) {
      Bs[cur ^ 1][kp][nn + 0] = (v2f){n0r.x, n1r.x};
      Bs[cur ^ 1][kp][nn + 1] = (v2f){n0r.y, n1r.y};
    }
    wait_async0();
    __syncthreads();
  }

  const int row = m0 + wm * 16 + hi * 8;
  const int c0 = n0 + wn * 16 + laneM;
  const int c1 = c0 + 16;
#pragma unroll
  for (int r = 0; r < 8; ++r) {
    O[(size_t)(row + r) * OUTW_ + C_ + c0] = acc0[r];
    O[(size_t)(row + r) * OUTW_ + C_ + c1] = acc1[r];
  }
}

// ---------------------------------------------------------------------------
// Kernel 5: out[b, t, 0:512] = x  (float4 copy)
// ---------------------------------------------------------------------------
__global__ __launch_bounds__(256) void concat_x_kernel(const float* __restrict__ x,
                                                       float* __restrict__ out) {
  const size_t idx = (size_t)blockIdx.x * 256 + threadIdx.x;  // float4 index
  const size_t row = idx >> 7;   // 128 float4 per 512-float row
  const size_t c4 = idx & 127;
  f4 val = *(const f4*)(x + (row << 9) + (c4 << 2));
  *(f4*)(out + row * OUTW_ + (c4 << 2)) = val;
}

// ---------------------------------------------------------------------------
extern "C" void kernel_launch(void* const* d_in, const int* in_sizes, int n_in,
                              void* d_out, int out_size, void* d_ws, size_t ws_size,
                              hipStream_t stream) {
  (void)in_sizes; (void)n_in; (void)out_size; (void)ws_size;

  const float* x  = (const float*)d_in[0];
  const float* Wq = (const float*)d_in[1];
  const float* bq = (const float*)d_in[2];
  const float* Wk = (const float*)d_in[3];
  const float* bk = (const float*)d_in[4];
  const float* Wv = (const float*)d_in[5];
  const float* bv = (const float*)d_in[6];
  float* out = (float*)d_out;
  float* ws = (float*)d_ws;

  const size_t NPROJ = (size_t)B_ * T_ * K_;   // 8,388,608 floats
  float* q      = ws;
  float* k      = ws + NPROJ;
  float* v      = ws + 2 * NPROJ;
  float* scores = ws + 3 * NPROJ;              // 8*T*T floats

  qkv_proj_kernel<<<dim3((B_ * T_) / 64, K_ / 64, 3), 256, 0, stream>>>(
      x, Wq, bq, Wk, bk, Wv, bv, ws);
  scores_kernel<<<dim3(T_ / 64, T_ / 64, B_), 256, 0, stream>>>(q, k, scores);
  softmax_q_kernel<<<dim3(T_ / 64, B_), 256, 0, stream>>>(scores);
  attn_kernel<<<dim3(T_ / 64, K_ / 64, B_), 256, 0, stream>>>(scores, v, out);
  concat_x_kernel<<<(B_ * T_ * C_ / 4) / 256, 256, 0, stream>>>(x, out);
}